// CapsuleConv2d_1460288880648
// MI455X (gfx1250) — compile-verified
//
#include <hip/hip_runtime.h>

typedef __attribute__((ext_vector_type(2))) float v2f;
typedef __attribute__((ext_vector_type(8))) float v8f;

#define HH 56
#define WW 56
#define HW 3136
#define TILES_PER_IMG 196      // 3136 / 16
#define WAVES_PER_BLOCK 4
#define NUM_TASKS (4 * 4 * TILES_PER_IMG)   // N * P_out * tiles = 3136

// Butterfly sum over each 16-lane half of the wave32 (xor masks 1,2,4,8 never
// cross lane bit 4). Result is replicated in every lane of the half.
__device__ __forceinline__ float hsum16(float v) {
#pragma unroll
  for (int m = 1; m < 16; m <<= 1) v += __shfl_xor(v, m, 32);
  return v;
}

// One wave = one (n, o, tile-of-16-spatial) task. Loops p=0..3; per p:
//   priors[k](16m x 16d) = two V_WMMA_F32_16X16X4_F32 (L_in = 8 = 2*K4),
//   then 3 routing iterations: softmax fully in-lane (logits replicated per
//   lane), only squash + logits-update use cross-lane butterflies.
__global__ __launch_bounds__(WAVES_PER_BLOCK * 32)
void caps_conv_route(const float* __restrict__ x, const float* __restrict__ wgt,
                     float* __restrict__ out) {
  __shared__ float lds[WAVES_PER_BLOCK][16][17];

  const int lane = threadIdx.x & 31;
  const int wv   = threadIdx.x >> 5;
  const int task = blockIdx.x * WAVES_PER_BLOCK + wv;
  const int tile = task % TILES_PER_IMG;
  const int o    = (task / TILES_PER_IMG) & 3;
  const int n    = task / (TILES_PER_IMG * 4);
  const int half = lane >> 4;       // which 16-lane half
  const int col  = lane & 15;       // A: row m ; B/C/D: column d
  const int flat0 = tile * 16;      // 16 consecutive flattened (h,w) positions

  // A-operand coordinates for this lane (A layout: M = lane%16, Kslot base = 2*half)
  const int fA = flat0 + col;
  const int hA = fA / WW;
  const int wA = fA - hA * WW;

  float out_acc[8];
#pragma unroll
  for (int j = 0; j < 8; ++j) out_acc[j] = 0.f;

  for (int p = 0; p < 4; ++p) {
    // ---------------- priors via fp32 WMMA ----------------
    v8f pri[9];
#pragma unroll
    for (int k = 0; k < 9; ++k) {
      const int dj = k / 3, dk = k % 3;
      const int y  = hA + dj - 1;
      const int xx = wA + dk - 1;
      // branchless padding: clamp address, zero via mask (keeps EXEC all-ones)
      const float msk = (y >= 0 && y < HH && xx >= 0 && xx < WW) ? 1.f : 0.f;
      const int yc = min(max(y, 0), HH - 1);
      const int xc = min(max(xx, 0), WW - 1);
      v8f c = {};
#pragma unroll
      for (int hl = 0; hl < 2; ++hl) {
        const int lb = hl * 4 + 2 * half;  // first of two L_in values this lane holds
        // A: patches[n, p, lb + {0,1}, yc, xc]
        const float* xp = x + (((size_t)(n * 4 + p) * 8 + lb) * HH + yc) * WW + xc;
        v2f a;
        a.x = msk * xp[0];
        a.y = msk * xp[HW];
        // B: weight[o, p, k, lb + {0,1}, col]
        const float* wp = wgt + (((size_t)((o * 4 + p) * 9 + k) * 8 + lb) * 16) + col;
        v2f b;
        b.x = wp[0];
        b.y = wp[16];
        c = __builtin_amdgcn_wmma_f32_16x16x4_f32(false, a, false, b, (short)0, c,
                                                  false, false);
      }
      pri[k] = c;
    }

    // ---------------- dynamic routing (3 iters) ----------------
    // C/D layout: lane holds d = col for spatial m = j + 8*half, j = VGPR index.
    // Each spatial position routes independently; logits replicated per lane.
#pragma unroll
    for (int j = 0; j < 8; ++j) {
      // ---- iteration 0 peeled: softmax(all-zero logits) == 1/9 exactly ----
      float s = 0.f;
#pragma unroll
      for (int k = 0; k < 9; ++k) s += pri[k][j];
      s *= (1.f / 9.f);
      float sq    = hsum16(s * s);
      float scale = (sq * __frcp_rn(1.f + sq)) * __frsqrt_rn(sq);
      float outv  = scale * s;

      // logits after iter 0 (initial logits were zero); butterfly replicates
      // the sum over d into every lane of the half.
      float lg[9];
#pragma unroll
      for (int k = 0; k < 9; ++k) lg[k] = hsum16(pri[k][j] * outv);

      // ---- iterations 1..2: softmax entirely in-lane ----
#pragma unroll
      for (int it = 1; it < 3; ++it) {
        float mx = lg[0];
#pragma unroll
        for (int k = 1; k < 9; ++k) mx = fmaxf(mx, lg[k]);
        float den = 0.f, acc = 0.f;
#pragma unroll
        for (int k = 0; k < 9; ++k) {
          const float e = __expf(lg[k] - mx);
          den += e;
          acc += e * pri[k][j];
        }
        s     = acc * __frcp_rn(den);
        sq    = hsum16(s * s);
        scale = (sq * __frcp_rn(1.f + sq)) * __frsqrt_rn(sq);
        outv  = scale * s;
        if (it == 1) {
#pragma unroll
          for (int k = 0; k < 9; ++k) lg[k] += hsum16(pri[k][j] * outv);
        }
      }
      out_acc[j] += outv;   // sum over input capsule planes p
    }
  }

  // ---------------- LDS transpose -> coalesced stores ----------------
  // registers hold (m = j+8*half, d = col); memory wants contiguous m per channel d.
#pragma unroll
  for (int j = 0; j < 8; ++j) lds[wv][col][j + 8 * half] = out_acc[j];
  // same-wave LDS ops are in-order; compiler inserts the dscnt wait on use.
#pragma unroll
  for (int r = 0; r < 8; ++r) {
    const int d = r + 8 * half;
    const float v = lds[wv][d][col];
    // out[n, o*16 + d, h, w] with flat spatial index flat0+col (contiguous per half)
    out[((size_t)((n * 4 + o) * 16 + d)) * HW + flat0 + col] = v;
  }
}

extern "C" void kernel_launch(void* const* d_in, const int* in_sizes, int n_in,
                              void* d_out, int out_size, void* d_ws, size_t ws_size,
                              hipStream_t stream) {
  const float* x   = (const float*)d_in[0];   // (4, 32, 56, 56)
  const float* wgt = (const float*)d_in[1];   // (4, 4, 9, 8, 16)
  float* out = (float*)d_out;                 // (4, 64, 56, 56)
  (void)in_sizes; (void)n_in; (void)out_size; (void)d_ws; (void)ws_size;

  dim3 grid(NUM_TASKS / WAVES_PER_BLOCK);     // 784 blocks
  dim3 block(WAVES_PER_BLOCK * 32);           // 4 waves
  hipLaunchKernelGGL(caps_conv_route, grid, block, 0, stream, x, wgt, out);
}